// Poolinglabel_91104846282958
// MI455X (gfx1250) — compile-verified
//
#include <hip/hip_runtime.h>
#include <stdint.h>

// Tile geometry: each workgroup produces an 8x32 tile of one (b) output plane.
#define TI 8
#define TJ 32
#define ROWS (4 * TI + 35)   // 67  input rows needed  (window [4i-19, 4i+19])
#define COLS (4 * TJ + 35)   // 163 input cols needed
#define NPIX (ROWS * COLS)   // 10921
#define H 512
#define W 512
#define OH 128
#define OW 128
#define NC 19

typedef __attribute__((address_space(1))) int* as1_ip;
typedef __attribute__((address_space(3))) int* as3_ip;

__global__ __launch_bounds__(256) void pool_label_kernel(const float* __restrict__ xg,
                                                         float* __restrict__ outg) {
    // Bitmask tile of the input region (1u << class per pixel), then row buffer
    // of vertically OR-reduced windows.
    __shared__ uint32_t mt[NPIX];          // 43,684 B
    __shared__ uint32_t vt[TI * COLS];     //  5,216 B

    const int blk = blockIdx.x;            // 8 batches * 16 i-tiles * 4 j-tiles = 512
    const int b   = blk >> 6;
    const int rem = blk & 63;
    const int i0  = (rem >> 2) * TI;       // 16 i-tiles of 8
    const int j0  = (rem & 3) * TJ;        // 4 j-tiles of 32
    const int y0  = 4 * i0 - 19;
    const int x0  = 4 * j0 - 19;
    const int t   = threadIdx.x;

    const float* xb = xg + (size_t)b * (H * W);

    // ---- Phase 1: stage the input region into LDS and convert to bitmasks ----
    // Out-of-range logical pixels read a clamped (valid) address and are then
    // overwritten with 0 (the OR identity) during conversion.
#if __has_builtin(__builtin_amdgcn_global_load_async_to_lds_b32)
    // CDNA5 async data mover: global -> LDS without a VGPR round trip
    // (GLOBAL_LOAD_ASYNC_TO_LDS_B32, tracked by ASYNCcnt).
    for (int k = t; k < NPIX; k += 256) {
        const int r  = k / COLS;
        const int c  = k - r * COLS;
        const int y  = y0 + r;
        const int x  = x0 + c;
        const int yc = min(max(y, 0), H - 1);
        const int xc = min(max(x, 0), W - 1);
        __builtin_amdgcn_global_load_async_to_lds_b32(
            (as1_ip)(xb + (size_t)yc * W + xc), (as3_ip)(&mt[k]), 0, 0);
    }
    asm volatile("s_wait_asynccnt 0" ::: "memory");  // wave's async copies done
    for (int k = t; k < NPIX; k += 256) {            // convert fp32 class -> bit
        const int r = k / COLS;
        const int c = k - r * COLS;
        const int y = y0 + r;
        const int x = x0 + c;
        const bool valid = (y >= 0) & (y < H) & (x >= 0) & (x < W);
        const float f = __uint_as_float(mt[k]);      // raw staged fp32 bits
        mt[k] = valid ? (1u << (int)f) : 0u;
    }
#else
    for (int k = t; k < NPIX; k += 256) {
        const int r  = k / COLS;
        const int c  = k - r * COLS;
        const int y  = y0 + r;
        const int x  = x0 + c;
        const int yc = min(max(y, 0), H - 1);
        const int xc = min(max(x, 0), W - 1);
        const bool valid = (y >= 0) & (y < H) & (x >= 0) & (x < W);
        const float f = xb[(size_t)yc * W + xc];
        mt[k] = valid ? (1u << (int)f) : 0u;
    }
#endif
    __syncthreads();

    // ---- Phase 2: vertical OR over 39 rows, window start stride 4 ----
    for (int k = t; k < TI * COLS; k += 256) {
        const int ti = k / COLS;
        const int c  = k - ti * COLS;
        const uint32_t* col = &mt[(4 * ti) * COLS + c];
        uint32_t acc = 0;
#pragma unroll
        for (int r = 0; r < 39; ++r) acc |= col[r * COLS];
        vt[k] = acc;
    }
    __syncthreads();

    // ---- Phase 3: horizontal OR over 39 cols + expand 19 channels ----
    {
        const int ti = t >> 5;   // wave-uniform row within tile
        const int tj = t & 31;   // lane = output column within tile
        const uint32_t* row = &vt[ti * COLS + 4 * tj];
        uint32_t h = 0;
#pragma unroll
        for (int c = 0; c < 39; ++c) h |= row[c];
        const int oi = i0 + ti;
        const int oj = j0 + tj;
        const size_t base = (((size_t)b * NC) * OH + oi) * OW + oj;
#pragma unroll
        for (int cc = 0; cc < NC; ++cc) {
            outg[base + (size_t)cc * (OH * OW)] = ((h >> cc) & 1u) ? 1.0f : 0.0f;
        }
    }
}

extern "C" void kernel_launch(void* const* d_in, const int* in_sizes, int n_in,
                              void* d_out, int out_size, void* d_ws, size_t ws_size,
                              hipStream_t stream) {
    (void)in_sizes; (void)n_in; (void)d_ws; (void)ws_size; (void)out_size;
    const float* x = (const float*)d_in[0];   // [8,512,512] fp32 class ids
    float* out = (float*)d_out;               // [8,19,128,128]
    pool_label_kernel<<<dim3(8 * 16 * 4), dim3(256), 0, stream>>>(x, out);
}